// PoissonEquation_17987323036039
// MI455X (gfx1250) — compile-verified
//
#include <hip/hip_runtime.h>
#include <stdint.h>

// ---------------------------------------------------------------------------
// Poisson weighted-Jacobi (omega=0.5) 5-point stencil, N=1024, 200 iterations.
// MI455X strategy: memory-bound problem, everything L2-resident (12MB << 192MB).
// Use CDNA5 async Global->LDS copies (ASYNCcnt) to stage row tiles, compute the
// stencil out of LDS, ping-pong d_out <-> d_ws across 200 launches.
// d_ws requirement: N*N*4 = 4 MiB.
// ---------------------------------------------------------------------------

#define NN      1024
#define ROWS    4              // interior rows computed per block
#define LROWS   (ROWS + 2)     // staged rows incl. halo
#define AS1     __attribute__((address_space(1)))
#define AS3     __attribute__((address_space(3)))

#ifndef __has_builtin
#define __has_builtin(x) 0
#endif

typedef int v4i_t __attribute__((ext_vector_type(4)));

// Per-lane async copy of 16 bytes from global memory into LDS (ASYNCcnt path).
__device__ __forceinline__ void async_ld_b128(AS3 float* lds, const float* gptr) {
#if __has_builtin(__builtin_amdgcn_global_load_async_to_lds_b128)
    __builtin_amdgcn_global_load_async_to_lds_b128(
        (AS1 v4i_t*)gptr, (AS3 v4i_t*)lds, 0, 0);
#else
    unsigned loff = (unsigned)(uintptr_t)lds;    // LDS byte address in a VGPR
    asm volatile("global_load_async_to_lds_b128 %0, %1, off"
                 :: "v"(loff), "v"(gptr) : "memory");
#endif
}

__device__ __forceinline__ void wait_async0() {
#if __has_builtin(__builtin_amdgcn_s_wait_asynccnt)
    __builtin_amdgcn_s_wait_asynccnt(0);
#else
    asm volatile("s_wait_asynccnt 0" ::: "memory");
#endif
}

// new = (1-w)*c + w*0.25*(up+dn+l+r - dx^2 * f), w = 0.5
__device__ __forceinline__ float upd(float c, float u, float d, float l, float r,
                                     float fv) {
    constexpr float DX2 = (float)((1.0 / 1023.0) * (1.0 / 1023.0));
    float s = (u + d) + (l + r);
    return 0.5f * c + 0.125f * (s - DX2 * fv);
}

__global__ __launch_bounds__(256)
void poisson_sor_step(const float* __restrict__ usrc,
                      const float* __restrict__ f,
                      float*       __restrict__ udst) {
    __shared__ float tile[LROWS * NN];            // 24 KB of the 320 KB WGP LDS

    const int t  = threadIdx.x;                   // 0..255
    const int c0 = t << 2;                        // 4 columns per thread
    const int r0 = 1 + (int)blockIdx.x * ROWS;    // first interior row of tile
    const int nrows = min(ROWS, (NN - 1) - r0);   // interior rows are 1..1022

    // ---- Stage LROWS rows of u into LDS via async Global->LDS b128 copies ----
    AS3 float* t3 = (AS3 float*)tile;
#pragma unroll
    for (int k = 0; k < LROWS; ++k) {
        int g = r0 - 1 + k;
        if (g > NN - 1) g = NN - 1;               // clamp for the last ragged tile
        async_ld_b128(t3 + k * NN + c0,
                      usrc + (size_t)g * NN + c0);
    }
    wait_async0();
    __syncthreads();

    // Horizontal-halo scalar indices (clamped; clamped lanes output 0 anyway).
    const int li = (c0 == 0)      ? 0        : (c0 - 1);
    const int ri = (c0 == NN - 4) ? (NN - 1) : (c0 + 4);

    // ---- Compute ROWS output rows from LDS, stream f in, store float4 out ----
    for (int rr = 0; rr < nrows; ++rr) {
        const int gr = r0 + rr;                   // global row
        const int k  = rr + 1;                    // LDS row of the center line

        const float4 c  = *(const float4*)&tile[k * NN + c0];
        const float4 up = *(const float4*)&tile[(k - 1) * NN + c0];
        const float4 dn = *(const float4*)&tile[(k + 1) * NN + c0];
        const float  lv = tile[k * NN + li];
        const float  rv = tile[k * NN + ri];
        const float4 fv = *(const float4*)&f[(size_t)gr * NN + c0];

        float4 o;
        o.x = (c0 == 0)      ? 0.0f : upd(c.x, up.x, dn.x, lv,  c.y, fv.x);
        o.y =                         upd(c.y, up.y, dn.y, c.x, c.z, fv.y);
        o.z =                         upd(c.z, up.z, dn.z, c.y, c.w, fv.z);
        o.w = (c0 == NN - 4) ? 0.0f : upd(c.w, up.w, dn.w, c.z, rv,  fv.w);

        *(float4*)&udst[(size_t)gr * NN + c0] = o;
    }
}

// Zero both ping-pong buffers (u0 = 0, and keeps Dirichlet boundary = 0, since
// the stencil kernel never writes boundary rows and stores 0 in boundary cols). Runs every call so the
// launch is deterministic regardless of prior buffer contents.
__global__ __launch_bounds__(256)
void poisson_zero2(float* __restrict__ a, float* __restrict__ b) {
    const size_t i = (size_t)blockIdx.x * blockDim.x + threadIdx.x;   // float4 idx
    const float4 z = make_float4(0.0f, 0.0f, 0.0f, 0.0f);
    ((float4*)a)[i] = z;
    ((float4*)b)[i] = z;
}

extern "C" void kernel_launch(void* const* d_in, const int* in_sizes, int n_in,
                              void* d_out, int out_size, void* d_ws, size_t ws_size,
                              hipStream_t stream) {
    const float* f   = (const float*)d_in[0];     // (1024,1024) f32
    float*       out = (float*)d_out;             // final u, also ping-pong buf A
    float*       wsb = (float*)d_ws;              // ping-pong buf B (4 MiB)

    // u0 = 0 in `out`; `wsb` zeroed too so its boundary stays 0 forever.
    poisson_zero2<<<(NN * NN / 4) / 256, 256, 0, stream>>>(out, wsb);

    const int nblocks = (NN - 2 + ROWS - 1) / ROWS;   // ceil(1022/4) = 256

    // Parity: it=0 reads out(zeros)->wsb ... it=199 (odd) reads wsb -> out.
    for (int it = 0; it < 200; ++it) {
        const float* src = (it & 1) ? wsb : out;
        float*       dst = (it & 1) ? out : wsb;
        poisson_sor_step<<<nblocks, 256, 0, stream>>>(src, f, dst);
    }
}